// PGATConvModel_76785425318521
// MI455X (gfx1250) — compile-verified
//
#include <hip/hip_runtime.h>
#include <math.h>

// ---------------- problem constants ----------------
#define BB   32
#define NN   512
#define FF   128
#define K4N  2048          // 4*N, MLP hidden
#define ALPHA   0.2f
#define THRESH  0.6f
#define NEG_INF (-9e15f)

typedef __attribute__((ext_vector_type(2))) float v2f;
typedef __attribute__((ext_vector_type(4))) float v4f;
typedef __attribute__((ext_vector_type(8))) float v8f;

__device__ __forceinline__ v8f vzero8() {
  v8f r;
#pragma unroll
  for (int i = 0; i < 8; ++i) r[i] = 0.f;
  return r;
}
__device__ __forceinline__ v2f mk2(float x, float y) {
  v2f r; r.x = x; r.y = y; return r;
}

// One f32 WMMA step: C(16x16) += A(16x4) * B(4x16)   (ISA 7.12.2 layouts)
//  A frag: lane holds A[row][k0 + 2*hi .. +1]          (row = lane&15, hi = lane>>4)
//  B frag: lane holds (B[hi][col], B[2+hi][col])       (col = lane&15)
//  D:      vgpr vi, lane -> (M = vi + 8*hi, N = col)
__device__ __forceinline__ v8f wmma_f32(v2f a, v2f b, v8f c) {
  return __builtin_amdgcn_wmma_f32_16x16x4_f32(false, a, false, b, (short)0, c,
                                               false, false);
}

// ============================================================
// Swizzle kernel: repack row-major [rows][rowlen4*4] into WMMA
// B-fragment pairs: dst[row][h][p] = (src[row][4p+h], src[row][4p+2+h])
// so a lane's B frag is one 8B load at (row=col, h=hi, p=k0/4).
// ============================================================
__global__ __launch_bounds__(256) void pgat_swz_kernel(
    const float* __restrict__ src, float* __restrict__ dst, int rowlen4,
    int total) {
  int idx = blockIdx.x * 256 + threadIdx.x;
  if (idx >= total) return;
  int row = idx / rowlen4;
  int p   = idx - row * rowlen4;
  v4f v = *(const v4f*)(src + ((size_t)row * rowlen4 + p) * 4);
  v2f* d = (v2f*)dst + (size_t)row * 2 * rowlen4;
  d[p]           = mk2(v.x, v.z);   // h = 0
  d[rowlen4 + p] = mk2(v.y, v.w);   // h = 1
}

// ============================================================
// Kernel 1:  Wh[r][c] = sum_f h[r][f] * W[c][f]
// Emits plain Wh (for the a-dot kernel) and WhSw (B-swizzled per
// batch: [b][f][h][p] pairs over node dim) for the final GEMM.
// grid 1024 (16 rows each), 256 thr = 8 waves (16 cols each)
// ============================================================
__global__ __launch_bounds__(256) void pgat_wh_kernel(
    const float* __restrict__ h, const float* __restrict__ Wsw,
    float* __restrict__ Wh, float* __restrict__ WhSw) {
  const int r0   = blockIdx.x * 16;
  const int c0   = (threadIdx.x >> 5) * 16;
  const int lane = threadIdx.x & 31;
  const int lrow = lane & 15;
  const int hi   = lane >> 4;

  const float* hrow = h + (size_t)(r0 + lrow) * FF + 2 * hi;
  const v2f*   bw   = (const v2f*)Wsw + ((size_t)(c0 + lrow) * 2 + hi) * (FF / 4);

  v8f acc = vzero8();
#pragma unroll 8
  for (int p = 0; p < FF / 4; ++p) {
    v2f a = *(const v2f*)(hrow + p * 4);
    acc = wmma_f32(a, bw[p], acc);
  }

  // plain Wh store (row r0 + vi + 8*hi, col c0+lrow)
  float* o = Wh + (size_t)(r0 + hi * 8) * FF + c0 + lrow;
#pragma unroll
  for (int vi = 0; vi < 8; ++vi) o[(size_t)vi * FF] = acc[vi];

  // swizzled store: lane owns rows nb+8hi .. +7 at col f -> pair groups
  const int bidx = r0 >> 9;          // batch
  const int nb   = r0 & (NN - 1);    // node base within batch
  const int pb   = (nb >> 2) + hi * 2;
  v2f* wsw = (v2f*)WhSw + ((size_t)bidx * FF + c0 + lrow) * 2 * (NN / 4);
  wsw[pb]                = mk2(acc[0], acc[2]);   // h=0, rows 4p+0,4p+2
  wsw[(NN / 4) + pb]     = mk2(acc[1], acc[3]);   // h=1, rows 4p+1,4p+3
  wsw[pb + 1]            = mk2(acc[4], acc[6]);
  wsw[(NN / 4) + pb + 1] = mk2(acc[5], acc[7]);
}

// ============================================================
// Kernel 2:  Wh1[r] = Wh[r,:] . a[:F],  Wh2[r] = Wh[r,:] . a[F:]
// one wave per row; grid 2048 x 256
// ============================================================
__global__ __launch_bounds__(256) void pgat_whab_kernel(
    const float* __restrict__ Wh, const float* __restrict__ a,
    float* __restrict__ Wh1, float* __restrict__ Wh2) {
  const int row  = blockIdx.x * 8 + (threadIdx.x >> 5);
  const int lane = threadIdx.x & 31;

  v4f v  = *(const v4f*)(Wh + (size_t)row * FF + lane * 4);
  v4f a1 = *(const v4f*)(a + lane * 4);
  v4f a2 = *(const v4f*)(a + FF + lane * 4);
  float s1 = v.x * a1.x + v.y * a1.y + v.z * a1.z + v.w * a1.w;
  float s2 = v.x * a2.x + v.y * a2.y + v.z * a2.z + v.w * a2.w;
#pragma unroll
  for (int off = 16; off > 0; off >>= 1) {
    s1 += __shfl_xor(s1, off, 32);
    s2 += __shfl_xor(s2, off, 32);
  }
  if (lane == 0) {
    Wh1[row] = s1;
    Wh2[row] = s2;
  }
}

// ============================================================
// Kernel 3 (mega): per 16-row tile of one batch:
//   E -> MLP (sigmoid, chunked K) -> +e, threshold/mask -> softmax
//   -> att @ Wh -> ELU -> out.  All B operands pre-swizzled.
// grid = B * N/16 = 1024, 256 thr = 8 waves
// ============================================================
#define ESTR  514          // LDS row stride for E / att (bank spread, 8B aligned)
#define TSTR  260          // LDS row stride for T chunk
#define CHUNK 256          // MLP hidden chunk width

__global__ __launch_bounds__(256) void pgat_main_kernel(
    const float* __restrict__ A0sw, const float* __restrict__ A2sw,
    const float* __restrict__ WhSw, const float* __restrict__ Wh1,
    const float* __restrict__ Wh2, float* __restrict__ out) {
  const int b    = blockIdx.x >> 5;          // batch
  const int n0   = (blockIdx.x & 31) * 16;   // row tile within batch
  const int tid  = threadIdx.x;
  const int wv   = tid >> 5;
  const int lane = tid & 31;
  const int lrow = lane & 15;
  const int hi   = lane >> 4;
  const int rowbase = b * NN + n0;           // flattened first row of tile

  __shared__ float sE[16 * ESTR];            // e tile, later att tile (32.9 KB)
  __shared__ float sT[16 * TSTR];            // sigmoid chunk (16.6 KB)

  // ---- build E = leakyrelu(Wh1[row] + Wh2[col]) ----
  for (int idx = tid; idx < 16 * NN; idx += 256) {
    int r = idx >> 9;
    int m = idx & (NN - 1);
    float v = Wh1[rowbase + r] + Wh2[b * NN + m];
    sE[r * ESTR + m] = v > 0.f ? v : ALPHA * v;
  }
  __syncthreads();

  // ---- adjacency MLP, chunked over hidden dim; adj accum in VGPRs ----
  v8f adj[4];
#pragma unroll
  for (int i = 0; i < 4; ++i) adj[i] = vzero8();

  const float* erow = sE + lrow * ESTR + 2 * hi;
  const float* trow = sT + lrow * TSTR + 2 * hi;
  // GEMM2 B base: m0(i) = wv*64 + i*16 ; constant stride between i's
  const v2f* b2base =
      (const v2f*)A2sw + ((size_t)(wv * 64 + lrow) * 2 + hi) * (K4N / 4);

  for (int kc = 0; kc < K4N; kc += CHUNK) {
    // GEMM1: sT[r][j-kc] = sigmoid( sum_m E[r][m] * A0[j][m] )
    // 2 j-tiles per wave, shared A-fragment.
    {
      const int j0 = kc + wv * 32;
      const v2f* b0 =
          (const v2f*)A0sw + ((size_t)(j0 + lrow) * 2 + hi) * (NN / 4);
      const v2f* b1 =
          (const v2f*)A0sw + ((size_t)(j0 + 16 + lrow) * 2 + hi) * (NN / 4);
      v8f t0 = vzero8(), t1 = vzero8();
#pragma unroll 4
      for (int p = 0; p < NN / 4; ++p) {
        v2f a = *(const v2f*)(erow + p * 4);
        t0 = wmma_f32(a, b0[p], t0);
        t1 = wmma_f32(a, b1[p], t1);
      }
      float* td0 = sT + (hi * 8) * TSTR + wv * 32 + lrow;
      float* td1 = td0 + 16;
#pragma unroll
      for (int vi = 0; vi < 8; ++vi) {
        td0[vi * TSTR] = 1.f / (1.f + __expf(-t0[vi]));
        td1[vi * TSTR] = 1.f / (1.f + __expf(-t1[vi]));
      }
    }
    __syncthreads();

    // GEMM2: adj[i] += Tchunk(16xCHUNK) @ A2 tiles ; shared A-fragment x4
    {
      const v2f* b2 = b2base + (kc >> 2);
#pragma unroll 2
      for (int p = 0; p < CHUNK / 4; ++p) {
        v2f a = *(const v2f*)(trow + p * 4);
#pragma unroll
        for (int i = 0; i < 4; ++i)
          adj[i] = wmma_f32(a, b2[(size_t)i * 16 * 2 * (K4N / 4) + p], adj[i]);
      }
    }
    __syncthreads();
  }

  // ---- adj = e + mlp; threshold+mask; write att in place over sE ----
  // (adj > THRESH) <=> (thresholded adj > 0) since THRESH > 0
#pragma unroll
  for (int i = 0; i < 4; ++i) {
    const int m = (wv * 4 + i) * 16 + lrow;
#pragma unroll
    for (int vi = 0; vi < 8; ++vi) {
      const int r = vi + hi * 8;
      float e = sE[r * ESTR + m];
      float av = e + adj[i][vi];
      sE[r * ESTR + m] = (av > THRESH) ? e : NEG_INF;
    }
  }
  __syncthreads();

  // ---- softmax over node dim: wave handles rows 2wv, 2wv+1 ----
#pragma unroll
  for (int rr = 0; rr < 2; ++rr) {
    float* row = sE + (wv * 2 + rr) * ESTR;
    float mx = -INFINITY;
    for (int m = lane; m < NN; m += 32) mx = fmaxf(mx, row[m]);
#pragma unroll
    for (int off = 16; off > 0; off >>= 1) mx = fmaxf(mx, __shfl_xor(mx, off, 32));
    float sum = 0.f;
    for (int m = lane; m < NN; m += 32) {
      float ev = __expf(row[m] - mx);   // NEG_INF entries underflow to 0
      row[m] = ev;
      sum += ev;
    }
#pragma unroll
    for (int off = 16; off > 0; off >>= 1) sum += __shfl_xor(sum, off, 32);
    float inv = 1.f / sum;
    for (int m = lane; m < NN; m += 32) row[m] *= inv;
  }
  __syncthreads();

  // ---- GEMM3: out = ELU( att(16xN) @ Wh_b(NxF) ); wave -> 16 feature cols ----
  {
    const int f0 = wv * 16;
    const v2f* bw =
        (const v2f*)WhSw + (((size_t)b * FF + f0 + lrow) * 2 + hi) * (NN / 4);
    v8f acc = vzero8();
#pragma unroll 8
    for (int p = 0; p < NN / 4; ++p) {
      v2f a = *(const v2f*)(erow + p * 4);   // erow == att row now
      acc = wmma_f32(a, bw[p], acc);
    }
    float* o = out + (size_t)(rowbase + hi * 8) * FF + f0 + lrow;
#pragma unroll
    for (int vi = 0; vi < 8; ++vi) {
      float x = acc[vi];
      o[(size_t)vi * FF] = x > 0.f ? x : (__expf(x) - 1.f);
    }
  }
}

// ============================================================
extern "C" void kernel_launch(void* const* d_in, const int* in_sizes, int n_in,
                              void* d_out, int out_size, void* d_ws,
                              size_t ws_size, hipStream_t stream) {
  const float* h  = (const float*)d_in[0];   // [B,N,F]
  const float* W  = (const float*)d_in[1];   // [F,F]
  const float* a  = (const float*)d_in[2];   // [2F,1]
  const float* A0 = (const float*)d_in[3];   // [4N,N]
  const float* A2 = (const float*)d_in[4];   // [N,4N]
  float* out = (float*)d_out;                // [B,N,F]

  // workspace layout (floats)
  float* Wh   = (float*)d_ws;                       // B*N*F
  float* Wh1  = Wh + (size_t)BB * NN * FF;          // B*N
  float* Wh2  = Wh1 + (size_t)BB * NN;              // B*N
  float* WhSw = Wh2 + (size_t)BB * NN;              // B*N*F (swizzled)
  float* A0sw = WhSw + (size_t)BB * NN * FF;        // 4N*N  (swizzled)
  float* A2sw = A0sw + (size_t)K4N * NN;            // N*4N  (swizzled)
  float* Wsw  = A2sw + (size_t)NN * K4N;            // F*F   (swizzled)

  // pre-swizzle all static B operands into WMMA B-fragment pairs
  pgat_swz_kernel<<<(K4N * (NN / 4) + 255) / 256, 256, 0, stream>>>(
      A0, A0sw, NN / 4, K4N * (NN / 4));
  pgat_swz_kernel<<<(NN * (K4N / 4) + 255) / 256, 256, 0, stream>>>(
      A2, A2sw, K4N / 4, NN * (K4N / 4));
  pgat_swz_kernel<<<(FF * (FF / 4) + 255) / 256, 256, 0, stream>>>(
      W, Wsw, FF / 4, FF * (FF / 4));

  pgat_wh_kernel<<<(BB * NN) / 16, 256, 0, stream>>>(h, Wsw, Wh, WhSw);
  pgat_whab_kernel<<<(BB * NN) / 8, 256, 0, stream>>>(Wh, a, Wh1, Wh2);
  pgat_main_kernel<<<(BB * NN) / 16, 256, 0, stream>>>(A0sw, A2sw, WhSw, Wh1,
                                                       Wh2, out);
}